// SelfAttention_9259949490206
// MI455X (gfx1250) — compile-verified
//
#include <hip/hip_runtime.h>
#include <math.h>

#define BB 8
#define LL 2048
#define DD 512          // DK == DV
#define TM 16           // query rows per workgroup
#define NW 8            // waves per workgroup
#define NT (NW * 32)    // 256 threads
#define SP (LL + 4)     // padded LDS row stride (floats) to spread banks

typedef float v2f __attribute__((ext_vector_type(2)));
typedef float v8f __attribute__((ext_vector_type(8)));

__global__ __launch_bounds__(NT) void SelfAttention_wmma_f32_kernel(
    const float* __restrict__ Q,
    const float* __restrict__ K,
    const float* __restrict__ V,
    const unsigned char* __restrict__ Mask,
    float* __restrict__ Out)
{
    // Scores / weights for 16 query rows over all 2048 keys (~128 KB)
    __shared__ float sS[TM * SP];
    __shared__ float sRed[TM * 16];
    __shared__ float sMax[TM];
    __shared__ float sSum[TM];

    const int wg   = blockIdx.x;            // 0 .. B*L/TM - 1
    const int b    = wg / (LL / TM);
    const int q0   = (wg % (LL / TM)) * TM; // first query row of this tile
    const int tid  = threadIdx.x;
    const int wave = tid >> 5;              // 0..7
    const int lane = tid & 31;

    const float* __restrict__ Qb = Q + (size_t)b * LL * DD;
    const float* __restrict__ Kb = K + (size_t)b * LL * DD;
    const float* __restrict__ Vb = V + (size_t)b * LL * DD;
    const unsigned char* __restrict__ mb = Mask + (size_t)b * LL;
    float* __restrict__ Ob = Out + (size_t)b * LL * DD;

    // Per-lane WMMA coordinates (wave32, 16x16 tiles)
    const int mrow = lane & 15;        // A: M index / B: N index / C: N index
    const int koff = (lane >> 4) * 2;  // A/B: K sub-offset per lane half
    const int mU   = (lane >> 4) * 8;  // C: M base for upper lane half

    // ---------------- Pass 1: logits = Q . K^T with exclusions ----------------
    // Wave handles key tiles t = wave, wave+8, ... (16 tiles each)
    const float* qrow = Qb + (size_t)(q0 + mrow) * DD + koff; // lane's A row
    for (int t = wave; t < LL / 16; t += NW) {
        const int n0 = t * 16;
        const float* krow = Kb + (size_t)(n0 + mrow) * DD + koff; // lane's B "row" (key)
        v8f acc = {0.f, 0.f, 0.f, 0.f, 0.f, 0.f, 0.f, 0.f};
#pragma unroll 4
        for (int kk = 0; kk < DD; kk += 4) {
            v2f a = *(const v2f*)(qrow + kk);  // Q[m][kk+koff .. +1]
            v2f bb = *(const v2f*)(krow + kk); // K[n][kk+koff .. +1]
            acc = __builtin_amdgcn_wmma_f32_16x16x4_f32(
                false, a, false, bb, (short)0, acc, false, false);
        }
        // Exclusions: masked key, diagonal, exact-zero score -> -inf
        const int gk = n0 + mrow;              // this lane's key column
        const bool kmask = (mb[gk] != 0);
#pragma unroll
        for (int r = 0; r < 8; ++r) {
            const int m  = r + mU;             // local query row
            const int gq = q0 + m;             // global query row
            float sv = acc[r];
            if (kmask || (gk == gq) || (sv == 0.0f)) sv = -INFINITY;
            sS[m * SP + gk] = sv;
        }
    }
    __syncthreads();

    // ---------------- Softmax (reference-faithful) ----------------
    {
        const int r  = tid >> 4;   // row 0..15
        const int c0 = tid & 15;   // 16 threads cooperate per row
        float mx = -INFINITY;
        for (int j = c0; j < LL; j += 16) mx = fmaxf(mx, sS[r * SP + j]);
        sRed[r * 16 + c0] = mx;
        __syncthreads();
        if (tid < TM) {
            float m2 = -INFINITY;
#pragma unroll
            for (int j = 0; j < 16; ++j) m2 = fmaxf(m2, sRed[tid * 16 + j]);
            sMax[tid] = (m2 == -INFINITY) ? 0.0f : m2;  // ref: where(isfinite(mx), mx, 0)
        }
        __syncthreads();
        const float rm = sMax[r];
        float ps = 0.f;
        for (int j = c0; j < LL; j += 16) {
            float e = expf(sS[r * SP + j] - rm);  // exp(-inf - rm) == 0
            sS[r * SP + j] = e;                   // keep unnormalized weights
            ps += e;
        }
        sRed[r * 16 + c0] = ps;
        __syncthreads();
        if (tid < TM) {
            float s = 0.f;
#pragma unroll
            for (int j = 0; j < 16; ++j) s += sRed[tid * 16 + j];
            sSum[tid] = (s == 0.f) ? 1.f : s;     // ref: where(s==0, 1, s)
        }
        __syncthreads();
    }

    // ---------------- Pass 2: Out = W . V ----------------
    // Wave owns 64 output columns: tiles [wave*64 + tt*16), tt = 0..3
    v8f oacc[4];
#pragma unroll
    for (int tt = 0; tt < 4; ++tt)
        oacc[tt] = (v8f){0.f, 0.f, 0.f, 0.f, 0.f, 0.f, 0.f, 0.f};

    const float* wrow = sS + mrow * SP + koff;  // A: W[m][kk+koff .. +1] from LDS
    const int colbase = wave * 64 + mrow;       // B/C: this lane's N per tile
    for (int kk = 0; kk < LL; kk += 4) {
        v2f a = *(const v2f*)(wrow + kk);
        const size_t krow = (size_t)(kk + koff) * DD;
#pragma unroll
        for (int tt = 0; tt < 4; ++tt) {
            const int col = colbase + tt * 16;
            v2f bb;
            bb.x = Vb[krow + col];        // V[kk+koff  ][col]
            bb.y = Vb[krow + DD + col];   // V[kk+koff+1][col]
            oacc[tt] = __builtin_amdgcn_wmma_f32_16x16x4_f32(
                false, a, false, bb, (short)0, oacc[tt], false, false);
        }
    }

    // Normalize, zero masked-query rows, store
#pragma unroll
    for (int r = 0; r < 8; ++r) {
        const int m  = r + mU;
        const int gq = q0 + m;
        const float inv = 1.0f / sSum[m];
        const bool qmask = (mb[gq] != 0);   // ref: masked query row -> all-zero output
#pragma unroll
        for (int tt = 0; tt < 4; ++tt) {
            const int col = colbase + tt * 16;
            float val = qmask ? 0.0f : (oacc[tt][r] * inv);
            Ob[(size_t)gq * DD + col] = val;
        }
    }
}

extern "C" void kernel_launch(void* const* d_in, const int* in_sizes, int n_in,
                              void* d_out, int out_size, void* d_ws, size_t ws_size,
                              hipStream_t stream) {
    (void)in_sizes; (void)n_in; (void)d_ws; (void)ws_size; (void)out_size;
    const float* Q = (const float*)d_in[0];
    const float* K = (const float*)d_in[1];
    const float* V = (const float*)d_in[2];
    const unsigned char* M = (const unsigned char*)d_in[3];
    float* O = (float*)d_out;

    dim3 grid(BB * (LL / TM));  // 1024 workgroups
    dim3 block(NT);             // 8 wave32
    SelfAttention_wmma_f32_kernel<<<grid, block, 0, stream>>>(Q, K, V, M, O);
}